// FastRGCNGNN_360777253370
// MI455X (gfx1250) — compile-verified
//
#include <hip/hip_runtime.h>

#define N_NODES   100000
#define N_EDGES   1600000
#define IN_CH     128
#define HIDDEN    64
#define N_REL     16
#define N_BASES   8
#define N_CLASSES 16

typedef float v2f __attribute__((ext_vector_type(2)));
typedef float v8f __attribute__((ext_vector_type(8)));

// ---------------- utility: zero a float region ----------------
__global__ void zero_f32(float* __restrict__ p, int n) {
    int i = blockIdx.x * blockDim.x + threadIdx.x;
    if (i < n) p[i] = 0.0f;
}

// ---------------- build per-relation weights (TRANSPOSED): W[17][O][K] ----------------
// r < 16 : W_r = sum_b comp[r,b] * basis[b];  r == 16 : root
// basis/root element (k, o) of relation r is stored at  r*per + o*K + k
// so that a WMMA B-fragment (two consecutive k at fixed o) is one b64 load.
template <int K, int OUTB>
__global__ void build_w(const float* __restrict__ basis, const float* __restrict__ comp,
                        const float* __restrict__ root, float* __restrict__ W) {
    const int per = K * OUTB;
    int idx = blockIdx.x * blockDim.x + threadIdx.x;
    if (idx >= 17 * per) return;
    int r = idx / per;
    int i = idx - r * per;          // i = k*OUTB + o  (source layout)
    int k = i / OUTB;
    int o = i - k * OUTB;
    float v;
    if (r == 16) {
        v = root[i];
    } else {
        v = 0.0f;
#pragma unroll
        for (int b = 0; b < N_BASES; ++b)
            v = fmaf(comp[r * N_BASES + b], basis[b * per + i], v);
    }
    W[r * per + o * K + k] = v;     // transposed store
}

// ---------------- fp32 WMMA GEMM:  C[M x N] = A[M x K] @ W ----------------
// W element (k, n):  (n/OUTB)*K*OUTB + (n%OUTB)*K + k   (per-block transposed)
// One wave computes a 16(M) x NT*16(N) strip: NT accumulators, A reused NTx.
// K-loop fully unrolled, branch-free; all addresses are base + immediate.
// Per K-step all NT+1 fragment loads are issued before the WMMA group so the
// scheduler can overlap load latency with the previous step's WMMAs.
// N-tail tiles clamp their load column to N-1 (valid memory, results dropped);
// only the stores are guarded (wave-uniform).
template <int NT, int K, int OUTB>
__global__ void __launch_bounds__(256)
rgcn_gemm_wmma(const float* __restrict__ A, const float* __restrict__ W,
               float* __restrict__ C, int M, int N) {
    const int lane  = threadIdx.x & 31;
    const int wave  = threadIdx.x >> 5;
    const int mTile = (blockIdx.y * 8 + wave) * 16;
    if (mTile >= M) return;                       // wave-uniform guard

    const int nTiles = N >> 4;
    const int nt0    = blockIdx.x * NT;
    const int l15    = lane & 15;
    const int hi     = lane >> 4;                 // 0: lanes 0-15, 1: lanes 16-31

    // A fragment (16x4 f32): lanes 0-15 -> K,K+1 ; lanes 16-31 -> K+2,K+3
    const int aOff = (mTile + l15) * K + hi * 2;

    // B fragment offsets (4x16 f32): lane -> column; hi selects K rows +2
    int wb[NT];
#pragma unroll
    for (int nt = 0; nt < NT; ++nt) {
        int col = (nt0 + nt) * 16 + l15;
        col = min(col, N - 1);                    // clamp tail tiles (loads only)
        wb[nt] = (col / OUTB) * (K * OUTB) + (col % OUTB) * K + hi * 2;
    }

    v8f acc[NT];
#pragma unroll
    for (int nt = 0; nt < NT; ++nt)
#pragma unroll
        for (int v = 0; v < 8; ++v) acc[nt][v] = 0.0f;

#pragma unroll
    for (int k = 0; k < K; k += 4) {
        // issue all fragment loads for this step up front (distinct SSA values)
        v2f a = *(const v2f*)(A + aOff + k);
        v2f b[NT];
#pragma unroll
        for (int nt = 0; nt < NT; ++nt)
            b[nt] = *(const v2f*)(W + wb[nt] + k);
#pragma unroll
        for (int nt = 0; nt < NT; ++nt)
            acc[nt] = __builtin_amdgcn_wmma_f32_16x16x4_f32(
                false, a, false, b[nt], (short)0, acc[nt], false, false);
    }

    // C/D layout: VGPR v, lanes 0-15 -> row v ; lanes 16-31 -> row v+8
#pragma unroll
    for (int nt = 0; nt < NT; ++nt) {
        if (nt0 + nt < nTiles) {                  // wave-uniform tail guard
            int col = (nt0 + nt) * 16 + l15;
#pragma unroll
            for (int v = 0; v < 8; ++v) {
                int row = mTile + hi * 8 + v;
                C[(size_t)row * N + col] = acc[nt][v];
            }
        }
    }
}

// ---------------- edge gather + atomic scatter ----------------
// msg_e[j] = Y[src_e, type_e*OD + j];  agg[dst_e, j] += msg; deg[dst_e] += 1
template <int OD>
__global__ void edge_scatter(const int* __restrict__ src, const int* __restrict__ dst,
                             const int* __restrict__ et, const float* __restrict__ Y,
                             float* __restrict__ agg, float* __restrict__ deg,
                             int nEdges, int doDeg) {
    long long t = (long long)blockIdx.x * blockDim.x + threadIdx.x;
    int e = (int)(t / OD);
    int j = (int)(t - (long long)e * OD);
    if (e >= nEdges) return;
    int s = src[e], d = dst[e], r = et[e];
    float m = Y[(size_t)s * (17 * OD) + r * OD + j];
    atomicAdd(agg + (size_t)d * OD + j, m);
    if (doDeg && j == 0) atomicAdd(deg + d, 1.0f);
}

// ---------------- layer-1 combine: h = relu(agg/deg + root + bias) ----------------
__global__ void combine_relu(const float* __restrict__ agg, const float* __restrict__ deg,
                             const float* __restrict__ Y, const float* __restrict__ bias,
                             float* __restrict__ h, int nNodes) {
    int idx = blockIdx.x * blockDim.x + threadIdx.x;
    if (idx >= nNodes * HIDDEN) return;
    int n = idx >> 6;
    int j = idx & 63;
    float dg = fmaxf(deg[n], 1.0f);
    float v  = agg[idx] / dg + Y[(size_t)n * (17 * HIDDEN) + 16 * HIDDEN + j] + bias[j];
    h[idx] = fmaxf(v, 0.0f);
}

// ---------------- layer-2 combine + log_softmax ----------------
__global__ void final_out(const float* __restrict__ agg, const float* __restrict__ deg,
                          const float* __restrict__ Y, const float* __restrict__ bias,
                          float* __restrict__ out, int nNodes) {
    int n = blockIdx.x * blockDim.x + threadIdx.x;
    if (n >= nNodes) return;
    float dg = fmaxf(deg[n], 1.0f);
    float lg[N_CLASSES];
    float mx = -3.402823466e38f;
#pragma unroll
    for (int j = 0; j < N_CLASSES; ++j) {
        float v = agg[n * N_CLASSES + j] / dg
                + Y[(size_t)n * (17 * N_CLASSES) + 16 * N_CLASSES + j] + bias[j];
        lg[j] = v;
        mx = fmaxf(mx, v);
    }
    float s = 0.0f;
#pragma unroll
    for (int j = 0; j < N_CLASSES; ++j) s += expf(lg[j] - mx);
    float lse = mx + logf(s);
#pragma unroll
    for (int j = 0; j < N_CLASSES; ++j) out[n * N_CLASSES + j] = lg[j] - lse;
}

// ---------------- host-side launch ----------------
extern "C" void kernel_launch(void* const* d_in, const int* in_sizes, int n_in,
                              void* d_out, int out_size, void* d_ws, size_t ws_size,
                              hipStream_t stream) {
    (void)in_sizes; (void)n_in; (void)out_size; (void)ws_size;

    const float* x      = (const float*)d_in[0];
    const int*   ei     = (const int*)d_in[1];
    const int*   etype  = (const int*)d_in[2];
    const float* basis1 = (const float*)d_in[3];
    const float* comp1  = (const float*)d_in[4];
    const float* root1  = (const float*)d_in[5];
    const float* bias1  = (const float*)d_in[6];
    const float* basis2 = (const float*)d_in[7];
    const float* comp2  = (const float*)d_in[8];
    const float* root2  = (const float*)d_in[9];
    const float* bias2  = (const float*)d_in[10];

    const int* src = ei;
    const int* dst = ei + N_EDGES;

    // workspace layout (floats)
    float* ws  = (float*)d_ws;
    float* W1  = ws;                                   // 17*128*64  = 139264
    float* W2  = W1 + 17 * IN_CH * HIDDEN;             // 17*64*16   = 17408
    float* y1  = W2 + 17 * HIDDEN * N_CLASSES;         // 100000*1088
    float* agg = y1 + (size_t)N_NODES * (17 * HIDDEN); // 100000*64 (later reused as 100000*16)
    float* deg = agg + (size_t)N_NODES * HIDDEN;       // 100000 (shared by both layers)
    float* h   = deg + N_NODES;                        // 100000*64
    float* y2  = y1;                                   // reuse y1 space (y1 dead after combine)
    float* out = (float*)d_out;

    const int N1 = 17 * HIDDEN;     // 1088
    const int N2 = 17 * N_CLASSES;  // 272

    // per-relation weight builds (tiny, writes transposed blocks)
    build_w<IN_CH, HIDDEN><<<(17 * IN_CH * HIDDEN + 255) / 256, 256, 0, stream>>>(
        basis1, comp1, root1, W1);
    build_w<HIDDEN, N_CLASSES><<<(17 * HIDDEN * N_CLASSES + 255) / 256, 256, 0, stream>>>(
        basis2, comp2, root2, W2);

    // zero agg1 + deg (contiguous: 64+1 floats per node)
    zero_f32<<<(N_NODES * 65 + 255) / 256, 256, 0, stream>>>(agg, N_NODES * 65);

    // layer 1 GEMM: y1 = x @ [W_0..W_15, root1]   (M=100000, K=128, N=1088)
    {
        dim3 g(N1 / 64, (N_NODES / 16 + 7) / 8);
        rgcn_gemm_wmma<4, IN_CH, HIDDEN>
            <<<g, 256, 0, stream>>>(x, W1, y1, N_NODES, N1);
    }
    // layer 1 edge gather/scatter (also builds deg)
    edge_scatter<HIDDEN><<<(int)(((long long)N_EDGES * HIDDEN + 255) / 256), 256, 0, stream>>>(
        src, dst, etype, y1, agg, deg, N_EDGES, 1);
    // h = relu(mean + x@root1 + bias1)
    combine_relu<<<(N_NODES * HIDDEN + 255) / 256, 256, 0, stream>>>(
        agg, deg, y1, bias1, h, N_NODES);

    // zero agg2 (reuses agg region; deg preserved)
    zero_f32<<<(N_NODES * N_CLASSES + 255) / 256, 256, 0, stream>>>(agg, N_NODES * N_CLASSES);

    // layer 2 GEMM: y2 = h @ [W'_0..W'_15, root2]  (M=100000, K=64, N=272)
    {
        dim3 g((N2 + 63) / 64, (N_NODES / 16 + 7) / 8);
        rgcn_gemm_wmma<4, HIDDEN, N_CLASSES>
            <<<g, 256, 0, stream>>>(h, W2, y2, N_NODES, N2);
    }
    // layer 2 edge gather/scatter (deg reused)
    edge_scatter<N_CLASSES><<<(int)(((long long)N_EDGES * N_CLASSES + 255) / 256), 256, 0, stream>>>(
        src, dst, etype, y2, agg, deg, N_EDGES, 0);

    // logits -> log_softmax
    final_out<<<(N_NODES + 255) / 256, 256, 0, stream>>>(agg, deg, y2, bias2, out, N_NODES);
}